// FusedMultiPool_2645699854881
// MI455X (gfx1250) — compile-verified
//
#include <hip/hip_runtime.h>

// Problem constants (match the reference).
constexpr int B        = 16;
constexpr int C        = 256;
constexpr int H        = 64;
constexpr int W        = 64;
constexpr int NUM_SETS = 64;
constexpr int SET_SIZE = 8;

constexpr int HW      = H * W;         // 4096 floats per channel plane (16 KB)
constexpr int HW4     = HW / 4;        // 1024 float4 per plane
constexpr int THREADS = 256;           // 8 wave32 waves per block
constexpr int ITERS   = HW4 / THREADS; // 4 float4 outputs per thread

// Native clang vector type: valid for __builtin_nontemporal_store and lowers
// to b128 global loads/stores.
typedef float v4f __attribute__((ext_vector_type(4)));

// One block per (batch, set) pair: reads 8 gathered channel planes,
// writes the per-element max plane.
__global__ __launch_bounds__(THREADS) void FusedMultiPool_gather_max8(
    const float* __restrict__ x,     // (B, C, H, W)
    const int*   __restrict__ idx,   // (NUM_SETS, SET_SIZE)
    float*       __restrict__ out)   // (B, NUM_SETS, H, W)
{
    const int bs = blockIdx.x;                 // 0 .. B*NUM_SETS-1
    const int b  = bs >> 6;                    // / NUM_SETS
    const int s  = bs & (NUM_SETS - 1);        // % NUM_SETS

    const float* xb = x + (size_t)b * C * HW;

    // Channel indices are uniform across the block: force them into SGPRs so
    // the 8 plane base addresses are scalar and loads use the saddr form.
    int c[SET_SIZE];
#pragma unroll
    for (int k = 0; k < SET_SIZE; ++k)
        c[k] = __builtin_amdgcn_readfirstlane(idx[s * SET_SIZE + k]);

    const v4f* p[SET_SIZE];
#pragma unroll
    for (int k = 0; k < SET_SIZE; ++k)
        p[k] = (const v4f*)(xb + (size_t)c[k] * HW);

    v4f* o = (v4f*)(out + (size_t)bs * HW);

#pragma unroll
    for (int it = 0; it < ITERS; ++it) {
        const int i = it * THREADS + threadIdx.x;   // float4 index in plane

        // 8 coalesced 128-bit loads (L2-resident x), reduce with max.
        v4f m = p[0][i];
#pragma unroll
        for (int k = 1; k < SET_SIZE; ++k) {
            const v4f v = p[k][i];
            m.x = fmaxf(m.x, v.x);
            m.y = fmaxf(m.y, v.y);
            m.z = fmaxf(m.z, v.z);
            m.w = fmaxf(m.w, v.w);
        }

        // Output is write-once / never re-read: non-temporal 128-bit store so
        // it doesn't evict the 64MB x working set from the 192MB L2.
        __builtin_nontemporal_store(m, &o[i]);
    }
}

extern "C" void kernel_launch(void* const* d_in, const int* in_sizes, int n_in,
                              void* d_out, int out_size, void* d_ws, size_t ws_size,
                              hipStream_t stream) {
    (void)in_sizes; (void)n_in; (void)out_size; (void)d_ws; (void)ws_size;

    const float* x   = (const float*)d_in[0];   // (16, 256, 64, 64) f32
    const int*   idx = (const int*)d_in[1];     // (64, 8) i32
    float*       out = (float*)d_out;           // (16, 64, 64, 64) f32

    dim3 grid(B * NUM_SETS);                    // 1024 blocks
    dim3 block(THREADS);                        // 256 threads = 8 waves
    FusedMultiPool_gather_max8<<<grid, block, 0, stream>>>(x, idx, out);
}